// BlockLinear_63428077027562
// MI455X (gfx1250) — compile-verified
//
#include <hip/hip_runtime.h>

typedef __attribute__((ext_vector_type(2))) float v2f;
typedef __attribute__((ext_vector_type(8))) float v8f;

#define BATCH_N   8192
#define IN_F      32768
#define OUT_F     2048
#define BLK       16          // in-features per output
#define OUT_TILES (OUT_F / 16)   // 128
#define WAVES_PER_BLOCK 8

// One wave32 computes a 16(batch) x 16(out) tile of y as
//   D = A(16x256) * B_blockdiag(256x16)
// via 64 chained V_WMMA_F32_16X16X4_F32 ops (exact fp32 FMA path).
__global__ __launch_bounds__(256) void block_linear_wmma_kernel(
    const float* __restrict__ x,   // (8192, 32768)
    const float* __restrict__ w,   // (2048, 16)
    float* __restrict__ y)         // (8192, 2048)
{
  const int lane = threadIdx.x & 31;
  const int wave = threadIdx.x >> 5;
  const int tile = blockIdx.x * WAVES_PER_BLOCK + wave;   // 0 .. 65535

  const int outTile   = tile % OUT_TILES;   // consecutive waves take adjacent out tiles (same batch rows)
  const int batchTile = tile / OUT_TILES;

  const int o0 = outTile * 16;
  const int b0 = batchTile * 16;
  const long c0 = (long)o0 * BLK;           // x column base: outputs o0..o0+15 consume cols c0..c0+255

  const int half = lane >> 4;               // which half-wave (selects K pair)
  const int lmod = lane & 15;               // A: row M ; B/D: column N

  // A fragment source: lane reads x[b0 + lmod, c0 + 4j + 2*half + {0,1}]  (8B aligned)
  const float* xrow = x + (size_t)(b0 + lmod) * IN_F + c0 + 2 * half;

  // B fragments: lane owns output column n = lmod -> weight row w[o0+lmod, :].
  // For chunk j it needs w[., (j&3)*4 + 2*half + {0,1}] -> 4 hoisted float2 loads.
  const float* wrow = w + (size_t)(o0 + lmod) * BLK + 2 * half;
  v2f wpair[4];
  #pragma unroll
  for (int q = 0; q < 4; ++q)
    wpair[q] = *(const v2f*)(wrow + 4 * q);

  const v2f bzero = {0.0f, 0.0f};
  v8f acc = {};   // C starts at zero

  #pragma unroll
  for (int j = 0; j < 64; ++j) {
    // A chunk j (16x4 fp32): lane layout M=lmod, K = 4j + 2*half + v   (ISA 7.12.2)
    v2f a = *(const v2f*)(xrow + 4 * j);
    // B chunk j (4x16): only column (j>>2) is non-zero (block-diagonal weight)
    v2f b = (lmod == (j >> 2)) ? wpair[j & 3] : bzero;
    // 8 args: (neg_a, A, neg_b, B, c_mod, C, reuse_a, reuse_b)
    acc = __builtin_amdgcn_wmma_f32_16x16x4_f32(
        false, a, false, b, (short)0, acc, false, false);
  }

  // D layout: lane -> column N=lmod ; VGPR r -> row M = r + 8*half
  float* ybase = y + (size_t)(b0 + 8 * half) * OUT_F + o0 + lmod;
  #pragma unroll
  for (int r = 0; r < 8; ++r)
    ybase[(size_t)r * OUT_F] = acc[r];
}

extern "C" void kernel_launch(void* const* d_in, const int* in_sizes, int n_in,
                              void* d_out, int out_size, void* d_ws, size_t ws_size,
                              hipStream_t stream) {
  (void)in_sizes; (void)n_in; (void)d_ws; (void)ws_size; (void)out_size;
  const float* x = (const float*)d_in[0];   // (8192, 32768) fp32
  const float* w = (const float*)d_in[1];   // (2048, 16)    fp32
  float* y = (float*)d_out;                 // (8192, 2048)  fp32

  const int totalTiles = (BATCH_N / 16) * OUT_TILES;          // 65536 waves
  const int blocks = totalTiles / WAVES_PER_BLOCK;            // 8192 blocks of 256 thr
  block_linear_wmma_kernel<<<blocks, 32 * WAVES_PER_BLOCK, 0, stream>>>(x, w, y);
}